// EdgeSIMPLEBatched_75187697484277
// MI455X (gfx1250) — compile-verified
//
#include <hip/hip_runtime.h>
#include <hip/hip_bf16.h>
#include <stdint.h>

#define NCOL 4096
#define MROW 128
#define KSEL 64
#define KP1  65
#define TENS 2
#define BDIM 64
#define EDIM 2
#define NEGC (-1.0e30f)
#define PFDEPTH 4

// ---------- CDNA5 async global->LDS (builtin confirmed present; wants int* params) ----------
#if __has_builtin(__builtin_amdgcn_global_load_async_to_lds_b32)
#define ASYNC_LOAD_F32(gsrc, ldst)                                              \
  __builtin_amdgcn_global_load_async_to_lds_b32(                                \
      (__attribute__((address_space(1))) int*)(uintptr_t)(gsrc),                \
      (__attribute__((address_space(3))) int*)(ldst), 0, 0)
#else
#define ASYNC_LOAD_F32(gsrc, ldst)                                              \
  do {                                                                          \
    unsigned _loff = (unsigned)(uintptr_t)(                                     \
        __attribute__((address_space(3))) void*)(ldst);                         \
    asm volatile("global_load_async_to_lds_b32 %0, %1, off"                     \
                 :: "v"(_loff), "v"(gsrc) : "memory");                          \
  } while (0)
#endif

#if __has_builtin(__builtin_amdgcn_s_wait_asynccnt)
#define WAIT_ASYNCCNT(n) __builtin_amdgcn_s_wait_asynccnt(n)
#else
#define WAIT_ASYNCCNT(n) asm volatile("s_wait_asynccnt %0" :: "i"(n) : "memory")
#endif

#if __has_builtin(__builtin_amdgcn_s_wait_dscnt)
#define WAIT_DSCNT0() __builtin_amdgcn_s_wait_dscnt(0)
#else
#define WAIT_DSCNT0() asm volatile("s_wait_dscnt 0" ::: "memory")
#endif

// ---------- math helpers ----------
__device__ __forceinline__ float lae(float a, float b) {  // logaddexp
  float m = fmaxf(a, b);
  float d = fabsf(a - b);
  return m + __logf(1.0f + __expf(-d));
}

// ESP carry striped over wave32: lane l holds c[l] (c0), c[32+l] (c1);
// c2 == c[64] is wave-uniform (all its inputs are broadcasts).
__device__ __forceinline__ void esp_step(float& c0, float& c1, float& c2,
                                         float th, int lane) {
  float s0 = __shfl_up(c0, 1, 32);     // c[l-1]
  float s1 = __shfl_up(c1, 1, 32);     // c[31+l]
  float w0 = __shfl(c0, 31, 32);       // c[31]
  float w1 = __shfl(c1, 31, 32);       // c[63]
  if (lane == 0) { s0 = NEGC; s1 = w0; }
  float n0 = lae(c0, s0 + th);
  float n1 = lae(c1, s1 + th);
  float n2 = lae(c2, w1 + th);
  c0 = n0; c1 = n1; c2 = n2;
}

// ---------- kernel (emitted first so --disasm snippet shows the async loop):
// forward prefix scan + exact-k marginals
// marg[m,n] = exp(th) * sum_i exp(P[n,i] + S[n+1,63-i] - log e_64)
__global__ void __launch_bounds__(32)
marginals_kernel(const float* __restrict__ scores, const float* __restrict__ Tab,
                 float* __restrict__ out_marg) {
  __shared__ float ring[PFDEPTH][64];
  const int m = blockIdx.x, lane = threadIdx.x;
  const int b = m >> 1, e = m & 1;
  const float* th_row = scores + (size_t)b * NCOL * EDIM + e;
  const float* Tm = Tab + (size_t)m * (NCOL + 1) * KP1;
  const float logEk = Tm[64];          // S(m, 0, 64) = log e_k(row)

  // async-prefetch suffix columns 1..PFDEPTH into the LDS ring
  for (int c = 1; c <= PFDEPTH; ++c) {
    const float* src = Tm + (size_t)c * KP1 + lane;
    float* dst = &ring[(c - 1) & (PFDEPTH - 1)][lane];
    ASYNC_LOAD_F32(src, dst);
    ASYNC_LOAD_F32(src + 32, dst + 32);
  }

  float p0 = (lane == 0) ? 0.0f : NEGC, p1 = NEGC, p2 = NEGC;
  float v = 0.0f;
  for (int n = 0; n < NCOL; ++n) {
    if ((n & 31) == 0) v = th_row[(size_t)(n + lane) * EDIM];
    float th = __shfl(v, n & 31, 32);

    // ensure column n+1 has landed (2 ops/column, completion is in-order)
    int rem = NCOL - n - 1;
    if (rem >= 3)      WAIT_ASYNCCNT(6);
    else if (rem == 2) WAIT_ASYNCCNT(4);
    else if (rem == 1) WAIT_ASYNCCNT(2);
    else               WAIT_ASYNCCNT(0);

    const int buf = n & (PFDEPTH - 1);
    float s_hi = ring[buf][63 - lane];   // pairs with i = lane
    float s_lo = ring[buf][31 - lane];   // pairs with i = lane + 32
    // exp-domain reduction: terms bounded by exp(-th) since the marginal <= 1
    float t = __expf(p0 + s_hi - logEk) + __expf(p1 + s_lo - logEk);
#pragma unroll
    for (int d = 1; d < 32; d <<= 1)     // wave32 sum reduction (plain adds)
      t += __shfl_xor(t, d, 32);
    if (lane == 0)
      out_marg[((size_t)b * NCOL + n) * EDIM + e] = __expf(th) * t;

    WAIT_DSCNT0();                       // ring reads done before buf reuse
    int c = n + 1 + PFDEPTH;             // maps back to this buf
    if (c <= NCOL) {
      const float* src = Tm + (size_t)c * KP1 + lane;
      float* dst = &ring[buf][lane];
      ASYNC_LOAD_F32(src, dst);
      ASYNC_LOAD_F32(src + 32, dst + 32);
    }
    esp_step(p0, p1, p2, th, lane);      // extend prefix with column n
  }
}

// ---------- kernel: suffix ESP table S(m, n, k), n = N..0 ----------
__global__ void __launch_bounds__(32)
suffix_table_kernel(const float* __restrict__ scores, float* __restrict__ Tab) {
  const int m = blockIdx.x;            // row 0..127
  const int lane = threadIdx.x;        // 0..31
  const int b = m >> 1, e = m & 1;
  const float* th_row = scores + (size_t)b * NCOL * EDIM + e;  // stride EDIM
  float* Tm = Tab + (size_t)m * (NCOL + 1) * KP1;

  float c0 = (lane == 0) ? 0.0f : NEGC;   // e_0 = 1 (log 0)
  float c1 = NEGC;
  float c2 = NEGC;
  {
    float* col = Tm + (size_t)NCOL * KP1;
    col[lane] = c0;
    col[32 + lane] = c1;
    if (lane == 0) col[64] = c2;
  }
  for (int n0 = NCOL - 32; n0 >= 0; n0 -= 32) {
    float v = th_row[(size_t)(n0 + lane) * EDIM];  // one chunk of theta, bcast via shfl
    for (int j = 31; j >= 0; --j) {
      float th = __shfl(v, j, 32);
      esp_step(c0, c1, c2, th, lane);
      float* col = Tm + (size_t)(n0 + j) * KP1;
      col[lane] = c0;
      col[32 + lane] = c1;
      if (lane == 0) col[64] = c2;
    }
  }
}

// ---------- kernel: conditional-Poisson sampling (256 independent chains) ----------
__global__ void __launch_bounds__(64)
sample_kernel(const float* __restrict__ scores, const float* __restrict__ u,
              const float* __restrict__ Tab, float* __restrict__ out_mask) {
  int gid = blockIdx.x * 64 + threadIdx.x;
  if (gid >= TENS * MROW) return;
  int t = gid >> 7;            // ensemble
  int m = gid & 127;
  int b = m >> 1, e = m & 1;
  const float* th_row = scores + (size_t)b * NCOL * EDIM + e;
  const float* um = u + ((size_t)t * MROW + m) * NCOL;
  const float* Tm = Tab + (size_t)m * (NCOL + 1) * KP1;
  float* om = out_mask + (size_t)t * BDIM * NCOL * EDIM;

  int r = KSEL;
  for (int n = 0; n < NCOL; ++n) {
    float p = 0.0f;
    if (r > 0) {
      float num = Tm[(size_t)(n + 1) * KP1 + (r - 1)];  // S(n+1, r-1)
      float den = Tm[(size_t)n * KP1 + r];              // S(n,   r)
      p = __expf(th_row[(size_t)n * EDIM] + num - den);
    }
    int inc = (um[n] < p) ? 1 : 0;
    om[((size_t)b * NCOL + n) * EDIM + e] = (float)inc;
    r -= inc;
  }
}

extern "C" void kernel_launch(void* const* d_in, const int* in_sizes, int n_in,
                              void* d_out, int out_size, void* d_ws, size_t ws_size,
                              hipStream_t stream) {
  (void)in_sizes; (void)n_in; (void)out_size; (void)ws_size;
  const float* scores = (const float*)d_in[0];   // (64, 4096, 2) f32
  const float* u      = (const float*)d_in[1];   // (2, 128, 4096) f32
  float* out      = (float*)d_out;
  float* out_mask = out;                                        // (2,64,4096,2)
  float* out_marg = out + (size_t)TENS * BDIM * NCOL * EDIM;    // (64,4096,2)
  float* Tab = (float*)d_ws;   // 128 * 4097 * 65 * 4B ~= 136.4 MB suffix table

  suffix_table_kernel<<<MROW, 32, 0, stream>>>(scores, Tab);
  marginals_kernel<<<MROW, 32, 0, stream>>>(scores, Tab, out_marg);
  sample_kernel<<<(TENS * MROW + 63) / 64, 64, 0, stream>>>(scores, u, Tab, out_mask);
}